// Net_60129542953
// MI455X (gfx1250) — compile-verified
//
#include <hip/hip_runtime.h>
#include <hip/hip_bf16.h>

// CDNA5 / gfx1250 implementation of a 2-layer GCN encoder + dot-product decoder.
// fp32 end-to-end; dense GEMMs use V_WMMA_F32_16X16X4_F32 (wave32).
// Each wave computes a 32x32 output tile (2x2 accumulators) so each A/B fragment
// feeds two WMMAs -> better wmma:vmem ratio than a single 16x16 tile per wave.

typedef __attribute__((ext_vector_type(2))) float v2f;
typedef __attribute__((ext_vector_type(8))) float v8f;

#define C_IN  128
#define C_HID 256
#define C_OUT 128

// ---------------------------------------------------------------- utility ----
__global__ void zero_f32(float* p, long long n) {
    long long i = (long long)blockIdx.x * blockDim.x + threadIdx.x;
    long long stride = (long long)gridDim.x * blockDim.x;
    for (; i < n; i += stride) p[i] = 0.0f;
}

__global__ void deg_count(const int* __restrict__ dst, float* __restrict__ deg, int E) {
    int e = blockIdx.x * blockDim.x + threadIdx.x;
    if (e < E) atomicAdd(&deg[dst[e]], 1.0f);
}

__global__ void deg_to_dinv(float* __restrict__ deg, int N) {
    int i = blockIdx.x * blockDim.x + threadIdx.x;
    if (i < N) deg[i] = rsqrtf(deg[i] + 1.0f);   // +1 self-loop
}

// ------------------------------------------------------------- WMMA GEMM ----
// C[M,Nout] = A[M,K] @ B[K,Nout] (+ bias[Nout] if non-null). Row-major.
// One wave computes a 32x32 tile as 2x2 16x16 WMMA accumulators; K stepped by 4.
// Requires M % 32 == 0 (N=100000 -> 3125 M-tiles) and Nout % 32 == 0.
// fp32 A-frag (16x4): lanes 0-15 rows M=0..15 hold K={k,k+1}; lanes 16-31 K={k+2,k+3}.
// fp32 B-frag (4x16): lanes 0-15 cols N=0..15 hold K={k,k+1}; lanes 16-31 K={k+2,k+3}.
// C/D frag: VGPR j -> M=j (lanes 0-15) / M=j+8 (lanes 16-31), N=lane&15.
__global__ __launch_bounds__(256) void gemm_wmma_f32(
    const float* __restrict__ A, const float* __restrict__ B,
    const float* __restrict__ bias, float* __restrict__ C,
    int M, int K, int Nout)
{
    int wave = (blockIdx.x * blockDim.x + threadIdx.x) >> 5;
    int lane = threadIdx.x & 31;
    int tilesN = Nout >> 5;                       // 32-wide N tiles
    int totalTiles = (M >> 5) * tilesN;
    if (wave >= totalTiles) return;               // wave-uniform; EXEC stays all-ones
    int mt = wave / tilesN;
    int nt = wave - mt * tilesN;
    int m0 = mt << 5, n0 = nt << 5;

    int row  = lane & 15;                         // M row (A) / N col (B,C)
    int ksel = (lane >> 4) << 1;                  // 0 for lanes 0-15, 2 for lanes 16-31

    const float* aptr0 = A + (size_t)(m0 + row) * K + ksel;
    const float* aptr1 = aptr0 + (size_t)16 * K;              // rows m0+16..m0+31
    const float* bptr0 = B + (size_t)ksel * Nout + n0 + row;
    const float* bptr1 = bptr0 + 16;                          // cols n0+16..n0+31

    v8f acc00 = {}, acc01 = {}, acc10 = {}, acc11 = {};
    for (int kk = 0; kk < K; kk += 4) {
        v2f a0 = *(const v2f*)(aptr0 + kk);
        v2f a1 = *(const v2f*)(aptr1 + kk);
        v2f b0, b1;
        b0.x = bptr0[(size_t)kk * Nout];
        b0.y = bptr0[(size_t)(kk + 1) * Nout];
        b1.x = bptr1[(size_t)kk * Nout];
        b1.y = bptr1[(size_t)(kk + 1) * Nout];
        acc00 = __builtin_amdgcn_wmma_f32_16x16x4_f32(false, a0, false, b0, (short)0, acc00, false, false);
        acc01 = __builtin_amdgcn_wmma_f32_16x16x4_f32(false, a0, false, b1, (short)0, acc01, false, false);
        acc10 = __builtin_amdgcn_wmma_f32_16x16x4_f32(false, a1, false, b0, (short)0, acc10, false, false);
        acc11 = __builtin_amdgcn_wmma_f32_16x16x4_f32(false, a1, false, b1, (short)0, acc11, false, false);
    }

    float bv0 = bias ? bias[n0 + row]      : 0.0f;
    float bv1 = bias ? bias[n0 + 16 + row] : 0.0f;
    int mbase = (lane >> 4) << 3;                 // 0 or 8
    float* c0 = C + (size_t)(m0 + mbase) * Nout + (n0 + row);
    float* c1 = c0 + (size_t)16 * Nout;
    #pragma unroll
    for (int j = 0; j < 8; ++j) {
        c0[(size_t)j * Nout]      = acc00[j] + bv0;
        c0[(size_t)j * Nout + 16] = acc01[j] + bv1;
        c1[(size_t)j * Nout]      = acc10[j] + bv0;
        c1[(size_t)j * Nout + 16] = acc11[j] + bv1;
    }
}

// -------------------------------------------------------- GCN aggregation ----
// agg[n][c] = h[n][c]*dinv[n]^2 + bias[c]   (self-loop term + conv bias)
__global__ void init_agg(const float* __restrict__ h, const float* __restrict__ dinv,
                         const float* __restrict__ bias, float* __restrict__ agg,
                         long long total, int Cshift)
{
    long long gid = (long long)blockIdx.x * blockDim.x + threadIdx.x;
    if (gid >= total) return;
    int C = 1 << Cshift;
    long long n = gid >> Cshift;
    int c = (int)(gid & (C - 1));
    float di = dinv[n];
    agg[gid] = h[gid] * di * di + bias[c];
}

// agg[dst][c..c+3] += h[src][c..c+3] * dinv[src]*dinv[dst]  over all edges.
// One thread handles 4 consecutive features: b128 load + 4 fp32 global atomics.
__global__ void scatter_edges_v4(const float* __restrict__ h,
                                 const int* __restrict__ src, const int* __restrict__ dst,
                                 const float* __restrict__ dinv, float* __restrict__ agg,
                                 int E, int Cshift)
{
    int qshift = Cshift - 2;                       // quads per row
    long long total = (long long)E << qshift;
    long long gid = (long long)blockIdx.x * blockDim.x + threadIdx.x;
    if (gid >= total) return;
    int e = (int)(gid >> qshift);
    int q = (int)(gid & ((1 << qshift) - 1));
    int s = src[e], d = dst[e];
    float coef = dinv[s] * dinv[d];
    const float4 hv = *(const float4*)(h + ((size_t)s << Cshift) + 4 * q);
    float* dest = agg + ((size_t)d << Cshift) + 4 * q;
    atomicAdd(dest + 0, hv.x * coef);
    atomicAdd(dest + 1, hv.y * coef);
    atomicAdd(dest + 2, hv.z * coef);
    atomicAdd(dest + 3, hv.w * coef);
}

// ------------------------------------------------------------- BatchNorm ----
// Thread f owns feature f; consecutive threads read consecutive addresses.
__global__ __launch_bounds__(256) void bn_stats(const float* __restrict__ x,
                                                float* __restrict__ sum,
                                                float* __restrict__ sumsq,
                                                int N, int rowsPerBlock)
{
    int f = threadIdx.x;                       // 0..255 (C_HID)
    int r0 = blockIdx.x * rowsPerBlock;
    int r1 = r0 + rowsPerBlock; if (r1 > N) r1 = N;
    float s = 0.0f, ss = 0.0f;
    for (int r = r0; r < r1; ++r) {
        float v = x[(size_t)r * C_HID + f];
        s += v; ss += v * v;
    }
    atomicAdd(&sum[f], s);
    atomicAdd(&sumsq[f], ss);
}

__global__ void bn_finalize(const float* __restrict__ sum, const float* __restrict__ sumsq,
                            const float* __restrict__ gamma, const float* __restrict__ beta,
                            float* __restrict__ scale, float* __restrict__ shift, int N)
{
    int f = threadIdx.x;
    float inv_n = 1.0f / (float)N;
    float mu = sum[f] * inv_n;
    float var = sumsq[f] * inv_n - mu * mu;    // biased variance
    float a = gamma[f] * rsqrtf(var + 1e-5f);
    scale[f] = a;
    shift[f] = beta[f] - mu * a;
}

__global__ void bn_apply_relu(const float* __restrict__ in,
                              const float* __restrict__ scale, const float* __restrict__ shift,
                              float* __restrict__ out, long long total)
{
    long long gid = (long long)blockIdx.x * blockDim.x + threadIdx.x;
    if (gid >= total) return;
    int f = (int)(gid & (C_HID - 1));
    float v = in[gid] * scale[f] + shift[f];
    out[gid] = v > 0.0f ? v : 0.0f;
}

// ------------------------------------------------------------- LayerNorm ----
__global__ __launch_bounds__(128) void ln_relu(const float* __restrict__ in,
                                               const float* __restrict__ gamma,
                                               const float* __restrict__ beta,
                                               float* __restrict__ out, int N)
{
    __shared__ float red[C_OUT];
    int n = blockIdx.x;
    int f = threadIdx.x;
    float v = in[(size_t)n * C_OUT + f];
    red[f] = v;
    __syncthreads();
    #pragma unroll
    for (int s = C_OUT / 2; s > 0; s >>= 1) {
        if (f < s) red[f] += red[f + s];
        __syncthreads();
    }
    float mu = red[0] * (1.0f / C_OUT);
    __syncthreads();
    float d = v - mu;
    red[f] = d * d;
    __syncthreads();
    #pragma unroll
    for (int s = C_OUT / 2; s > 0; s >>= 1) {
        if (f < s) red[f] += red[f + s];
        __syncthreads();
    }
    float var = red[0] * (1.0f / C_OUT);
    float r = d * rsqrtf(var + 1e-5f) * gamma[f] + beta[f];
    out[(size_t)n * C_OUT + f] = r > 0.0f ? r : 0.0f;
}

// ---------------------------------------------------------------- decode ----
// One wave32 per query pair: 128-dim dot, shuffle-reduce, write value + threshold.
__global__ __launch_bounds__(256) void decode(const float* __restrict__ z,
                                              const int* __restrict__ ia,
                                              const int* __restrict__ ib,
                                              const float* __restrict__ cutoff,
                                              float* __restrict__ out, int EL)
{
    int pair = (blockIdx.x * blockDim.x + threadIdx.x) >> 5;
    int lane = threadIdx.x & 31;
    if (pair >= EL) return;                     // wave-uniform
    const float* za = z + (size_t)ia[pair] * C_OUT;
    const float* zb = z + (size_t)ib[pair] * C_OUT;
    float dot = 0.0f;
    #pragma unroll
    for (int t = 0; t < C_OUT / 32; ++t)
        dot += za[lane + 32 * t] * zb[lane + 32 * t];
    #pragma unroll
    for (int off = 16; off > 0; off >>= 1)
        dot += __shfl_xor(dot, off, 32);        // wave32 reduce
    if (lane == 0) {
        out[pair] = dot;
        out[EL + pair] = (dot < cutoff[0]) ? 0.0f : 1.0f;
    }
}

// ------------------------------------------------------------------ host ----
static inline size_t align256(size_t x) { return (x + 255) & ~(size_t)255; }

extern "C" void kernel_launch(void* const* d_in, const int* in_sizes, int n_in,
                              void* d_out, int out_size, void* d_ws, size_t ws_size,
                              hipStream_t stream) {
    const float* x       = (const float*)d_in[0];
    const int*   eidx    = (const int*)d_in[1];
    const int*   elidx   = (const int*)d_in[2];
    const float* cutoff  = (const float*)d_in[3];
    const float* W1      = (const float*)d_in[4];
    const float* b1      = (const float*)d_in[5];
    const float* bn_g    = (const float*)d_in[6];
    const float* bn_b    = (const float*)d_in[7];
    const float* W2      = (const float*)d_in[8];
    const float* b2      = (const float*)d_in[9];
    const float* ln_g    = (const float*)d_in[10];
    const float* ln_b    = (const float*)d_in[11];
    const float* lin_W   = (const float*)d_in[12];
    const float* lin_b   = (const float*)d_in[13];

    const int N  = in_sizes[0] / C_IN;      // 100000 (divisible by 32)
    const int E  = in_sizes[1] / 2;         // 800000
    const int EL = in_sizes[2] / 2;         // 200000
    const int* src = eidx;
    const int* dst = eidx + E;
    const int* la  = elidx;
    const int* lb  = elidx + EL;

    // Workspace layout
    char* ws = (char*)d_ws;
    size_t off = 0;
    float* dinv   = (float*)(ws + off); off += align256((size_t)N * sizeof(float));
    float* bn_sum = (float*)(ws + off); off += align256(4 * C_HID * sizeof(float));
    float* bn_ssq = bn_sum + C_HID;
    float* bn_sc  = bn_sum + 2 * C_HID;
    float* bn_sh  = bn_sum + 3 * C_HID;
    float* bufA   = (float*)(ws + off); off += align256((size_t)N * C_HID * sizeof(float));
    float* bufB   = (float*)(ws + off);

    float* out = (float*)d_out;

    // 0) zero degree + BN accumulators
    zero_f32<<<(N + 255) / 256, 256, 0, stream>>>(dinv, (long long)N);
    zero_f32<<<2, 256, 0, stream>>>(bn_sum, 2 * C_HID);

    // 1) degree -> dinv
    deg_count<<<(E + 255) / 256, 256, 0, stream>>>(dst, dinv, E);
    deg_to_dinv<<<(N + 255) / 256, 256, 0, stream>>>(dinv, N);

    // 2) GEMM1: h1 = x @ W1  (bufA)
    {
        int tiles = (N / 32) * (C_HID / 32);      // 3125 * 8 = 25000 waves
        gemm_wmma_f32<<<(tiles + 7) / 8, 256, 0, stream>>>(x, W1, nullptr, bufA, N, C_IN, C_HID);
    }
    // 3) agg1 = h1*dinv^2 + b1 (bufB), then scatter edges
    {
        long long total = (long long)N * C_HID;
        init_agg<<<(int)((total + 255) / 256), 256, 0, stream>>>(bufA, dinv, b1, bufB, total, 8);
        long long work = (long long)E * (C_HID / 4);
        scatter_edges_v4<<<(int)((work + 255) / 256), 256, 0, stream>>>(bufA, src, dst, dinv, bufB, E, 8);
    }
    // 4) BatchNorm + ReLU -> bufA
    {
        int rpb = 256;
        bn_stats<<<(N + rpb - 1) / rpb, 256, 0, stream>>>(bufB, bn_sum, bn_ssq, N, rpb);
        bn_finalize<<<1, C_HID, 0, stream>>>(bn_sum, bn_ssq, bn_g, bn_b, bn_sc, bn_sh, N);
        long long total = (long long)N * C_HID;
        bn_apply_relu<<<(int)((total + 255) / 256), 256, 0, stream>>>(bufB, bn_sc, bn_sh, bufA, total);
    }
    // 5) GEMM2: h2 = hbn @ W2  (bufB)
    {
        int tiles = (N / 32) * (C_OUT / 32);      // 3125 * 4 = 12500 waves
        gemm_wmma_f32<<<(tiles + 7) / 8, 256, 0, stream>>>(bufA, W2, nullptr, bufB, N, C_HID, C_OUT);
    }
    // 6) agg2 = h2*dinv^2 + b2 (bufA), scatter
    {
        long long total = (long long)N * C_OUT;
        init_agg<<<(int)((total + 255) / 256), 256, 0, stream>>>(bufB, dinv, b2, bufA, total, 7);
        long long work = (long long)E * (C_OUT / 4);
        scatter_edges_v4<<<(int)((work + 255) / 256), 256, 0, stream>>>(bufB, src, dst, dinv, bufA, E, 7);
    }
    // 7) LayerNorm + ReLU -> bufB
    ln_relu<<<N, C_OUT, 0, stream>>>(bufA, ln_g, ln_b, bufB, N);

    // 8) GEMM3: z = hln @ lin_W + lin_b  (bufA)
    {
        int tiles = (N / 32) * (C_OUT / 32);
        gemm_wmma_f32<<<(tiles + 7) / 8, 256, 0, stream>>>(bufB, lin_W, lin_b, bufA, N, C_OUT, C_OUT);
    }
    // 9) decode: per-pair dot + threshold
    decode<<<(EL * 32 + 255) / 256, 256, 0, stream>>>(bufA, la, lb, cutoff, out, EL);
}